// CppFUSEModel_17386027614780
// MI455X (gfx1250) — compile-verified
//
#include <hip/hip_runtime.h>
#include <hip/hip_bf16.h>
#include <math.h>

// ---------------------------------------------------------------------------
// FUSE-style 2-reservoir hydrological model + gamma-kernel routing, CDNA5.
//   Phase 1 (serial scan, latency-bound): one wave32, state in VGPRs,
//            coalesced cooperative forcing loads + __shfl broadcast.
//   Phase 2 (routing FIR): Toeplitz conv mapped onto V_WMMA_F32_16X16X4_F32.
// ---------------------------------------------------------------------------

typedef __attribute__((ext_vector_type(2))) float v2f;
typedef __attribute__((ext_vector_type(8))) float v8f;

#define ROUTE_SHAPE 2.5f
#define KLEN 64

__device__ __forceinline__ float sigmoid_(float x) { return 1.0f / (1.0f + __expf(-x)); }
__device__ __forceinline__ float clamp_(float x, float lo, float hi) {
  return fminf(fmaxf(x, lo), hi);
}

// ---------------------------------------------------------------------------
// Phase 1: serial simulation on a single wave32 + gamma-kernel softmax epilogue
// ---------------------------------------------------------------------------
__global__ __launch_bounds__(32) void fuse_sim_kernel(
    const float* __restrict__ forcing,   // [T,3] (prcp, pet, temp)
    const float* __restrict__ raw,       // [29]
    float* __restrict__ runoff,          // [T]  (workspace)
    float* __restrict__ kern,            // [64] (workspace)
    int T)
{
  const int lane = threadIdx.x & 31;

  // physical params (low + (high-low)*sigmoid(raw)); each lane redundant.
  const float S1_max = 50.0f    + (5000.0f  - 50.0f)   * sigmoid_(raw[0]);
  const float S2_max = 100.0f   + (10000.0f - 100.0f)  * sigmoid_(raw[1]);
  const float f_tens = 0.05f    + (0.95f    - 0.05f)   * sigmoid_(raw[2]);
  const float ku     = 0.01f    + (1000.0f  - 0.01f)   * sigmoid_(raw[6]);
  const float c_exp  = 1.0f     + (20.0f    - 1.0f)    * sigmoid_(raw[7]);
  const float ki     = 0.01f    + (1000.0f  - 0.01f)   * sigmoid_(raw[11]);
  const float ks     = 0.001f   + (10000.0f - 0.001f)  * sigmoid_(raw[12]);
  const float n_exp  = 1.0f     + (10.0f    - 1.0f)    * sigmoid_(raw[13]);
  const float b_exp  = 0.001f   + (3.0f     - 0.001f)  * sigmoid_(raw[18]);
  const float mu_t   = 0.01f    + (5.0f     - 0.01f)   * sigmoid_(raw[21]);
  const float T_rain = -2.0f    + (4.0f     - (-2.0f)) * sigmoid_(raw[22]);

  const float invS1 = 1.0f / S1_max;
  const float invS2 = 1.0f / S2_max;
  const float invft = 1.0f / f_tens;

  float S1 = 50.0f, S2 = 250.0f;

  for (int c0 = 0; c0 < T; c0 += 32) {
    const int t = c0 + lane;
    // cooperative, coalesced load of 32 steps of forcing
    const float pr_v = forcing[3 * t + 0];
    const float pe_v = forcing[3 * t + 1];
    const float tm_v = forcing[3 * t + 2];
    if (c0 + 32 < T)  // hint next chunk into cache (global_prefetch_b8)
      __builtin_prefetch(&forcing[3 * (c0 + 32 + lane)], 0, 1);

    float my_runoff = 0.0f;
    #pragma unroll 8
    for (int j = 0; j < 32; ++j) {
      const float prcp = __shfl(pr_v, j);
      const float pet  = __shfl(pe_v, j);
      const float temp = __shfl(tm_v, j);

      const float rain = (temp > T_rain) ? prcp : 0.0f;
      const float s1f  = clamp_(S1 * invS1, 0.0f, 1.0f);
      const float s2f  = clamp_(S2 * invS2, 0.0f, 1.0f);

      const float l1  = __log2f(s1f);              // shared between b and c pows
      const float qsx = rain * exp2f(b_exp * l1);
      const float q12 = ku   * exp2f(c_exp * l1);
      const float e1  = pet * fminf(s1f * invft, 1.0f);
      const float qif = ki * s1f;
      const float qb  = ks * exp2f(n_exp * __log2f(s2f));

      S1 = clamp_(S1 + (rain - qsx - e1 - q12 - qif), 0.0f, S1_max);
      S2 = clamp_(S2 + (q12 - qb), 0.0f, S2_max);

      const float r = qsx + qif + qb;
      if (j == lane) my_runoff = r;                // lane j keeps step j
    }
    runoff[t] = my_runoff;                         // coalesced store
  }

  // ---- gamma routing kernel: softmax((shape-1)*ln(t) - t/theta), 64 taps ----
  const float delay = mu_t;                        // stop_gradient irrelevant fwd
  const float theta = fmaxf(delay, 1e-3f) / ROUTE_SHAPE;
  const float invth = 1.0f / theta;
  const float t0 = (float)lane + 0.5f;
  const float t1 = (float)lane + 32.5f;
  float a0 = (ROUTE_SHAPE - 1.0f) * __logf(t0) - t0 * invth;
  float a1 = (ROUTE_SHAPE - 1.0f) * __logf(t1) - t1 * invth;
  float mx = fmaxf(a0, a1);
  for (int off = 16; off > 0; off >>= 1) mx = fmaxf(mx, __shfl_xor(mx, off));
  float e0 = __expf(a0 - mx);
  float e1 = __expf(a1 - mx);
  float s = e0 + e1;
  for (int off = 16; off > 0; off >>= 1) s += __shfl_xor(s, off);
  const float inv = 1.0f / s;
  kern[lane]      = e0 * inv;
  kern[lane + 32] = e1 * inv;
}

// ---------------------------------------------------------------------------
// Phase 2: causal 64-tap FIR via V_WMMA_F32_16X16X4_F32.
// Per wave: 16 outputs out[base..base+15]; A = Toeplitz runoff window,
// B = tap weights broadcast across N; K=64 contracted as 16 chained WMMAs.
// ---------------------------------------------------------------------------
__global__ __launch_bounds__(256) void fuse_route_wmma_kernel(
    const float* __restrict__ runoff,    // [T]
    const float* __restrict__ kern,      // [64]
    float* __restrict__ out,             // [T]
    int T)
{
  __shared__ float stage[8][96];         // per-wave window x[base-64 .. base+32)

  const int lane = threadIdx.x & 31;
  const int wv   = threadIdx.x >> 5;     // wave id in block (8 waves)
  const int base = (blockIdx.x * 8 + wv) * 16;

  // stage the runoff window (zero-padded for t < 0)
  #pragma unroll
  for (int seg = 0; seg < 3; ++seg) {
    const int idx = base - 64 + lane + 32 * seg;
    stage[wv][lane + 32 * seg] = (idx >= 0 && idx < T) ? runoff[idx] : 0.0f;
  }
  __syncthreads();

  // A-matrix 16x4 f32 layout: lanes 0-15 hold M=lane, K={0,1} in vgpr{0,1};
  // lanes 16-31 hold M=lane-16, K={2,3}.  B 4x16: K per lane-half, bcast over N.
  const int m  = lane & 15;
  const int kk = (lane < 16) ? 0 : 2;

  v8f acc = {0.0f, 0.0f, 0.0f, 0.0f, 0.0f, 0.0f, 0.0f, 0.0f};

  #pragma unroll
  for (int p = 0; p < 16; ++p) {         // taps j = 4p + k, k in [0,4)
    v2f a, b;
    const int j = m - 4 * p - kk;        // x offset rel. base, in [-63, 15]
    a.x = stage[wv][64 + j];             // A[m][kk]   = x[base + m - (4p+kk)]
    a.y = stage[wv][63 + j];             // A[m][kk+1] = x[base + m - (4p+kk+1)]
    b.x = kern[4 * p + kk];              // B[kk][n]   = kern[4p+kk]   (all n)
    b.y = kern[4 * p + kk + 1];          // B[kk+1][n] = kern[4p+kk+1] (all n)
    acc = __builtin_amdgcn_wmma_f32_16x16x4_f32(
        /*neg_a=*/false, a, /*neg_b=*/false, b,
        /*c_mod=*/(short)0, acc, /*reuse_a=*/false, /*reuse_b=*/false);
  }

  // D[m][n] == out[base+m] for every n. C/D layout: lanes 0-15 -> M=v (N=lane),
  // lanes 16-31 -> M=v+8.  Write from the N==0 column (lanes 0 and 16).
  if (base < T && (lane & 15) == 0) {
    float* o = out + base + (lane >> 4) * 8;
    #pragma unroll
    for (int q = 0; q < 8; ++q) o[q] = acc[q];
  }
}

// ---------------------------------------------------------------------------
extern "C" void kernel_launch(void* const* d_in, const int* in_sizes, int n_in,
                              void* d_out, int out_size, void* d_ws, size_t ws_size,
                              hipStream_t stream) {
  const float* forcing = (const float*)d_in[0];   // [T,3] f32
  const float* raw     = (const float*)d_in[1];   // [29]  f32
  float* out = (float*)d_out;                     // [T]   f32

  const int T = in_sizes[0] / 3;

  float* runoff = (float*)d_ws;                   // [T]
  float* kern   = runoff + T;                     // [64]

  fuse_sim_kernel<<<1, 32, 0, stream>>>(forcing, raw, runoff, kern, T);

  const int tiles  = (T + 15) / 16;
  const int blocks = (tiles + 7) / 8;             // 8 waves (tiles) per block
  fuse_route_wmma_kernel<<<blocks, 256, 0, stream>>>(runoff, kern, out, T);
}